// MMCL_68667937128728
// MI455X (gfx1250) — compile-verified
//
#include <hip/hip_runtime.h>
#include <stdint.h>

#define BB      2048
#define NN      32768
#define KK      327          // int(0.01 * (NN - 1))
#define SCALEF  10.0f
#define NTHREADS 1024
#define EPT     (NN / NTHREADS)   // 32 elements per thread

// Order-preserving float -> uint key (ascending float <-> ascending uint)
__device__ __forceinline__ uint32_t f2key(float f) {
    uint32_t u = __float_as_uint(f);
    return (u & 0x80000000u) ? ~u : (u | 0x80000000u);
}
__device__ __forceinline__ float key2f(uint32_t k) {
    uint32_t u = (k & 0x80000000u) ? (k ^ 0x80000000u) : ~k;
    return __uint_as_float(u);
}

__global__ __launch_bounds__(NTHREADS, 1)
void hardneg_ce_rows(const float* __restrict__ logits,
                     const int*  __restrict__ targets,
                     float*      __restrict__ row_loss) {
    __shared__ float    rowData[NN];      // 128 KB: whole row in LDS
    __shared__ uint32_t hist[256];
    __shared__ float    red[NTHREADS];
    __shared__ uint32_t s_prefix;
    __shared__ uint32_t s_remaining;

    const int row    = blockIdx.x;
    const int tid    = threadIdx.x;
    const int target = targets[row];
    const char* gbase = (const char*)(logits + (size_t)row * NN);

    // ---------------- async global -> LDS copy of the row (ASYNCcnt path) ----
    // Each thread issues 8 x 16B async-to-LDS loads; addresses are per-lane VGPRs.
    uint32_t lds_base = (uint32_t)(uintptr_t)&rowData[0];
    #pragma unroll
    for (int j = 0; j < EPT / 4; ++j) {
        uint32_t off   = (uint32_t)(j * NTHREADS + tid) * 16u;
        uint32_t laddr = lds_base + off;
        const char* g  = gbase + off;
        asm volatile("global_load_async_to_lds_b128 %0, %1, off"
                     :: "v"(laddr), "v"(g) : "memory");
    }
    asm volatile("s_wait_asynccnt 0" ::: "memory");
    __syncthreads();

    // ---------------- 4-pass radix select: exact k-th largest key ------------
    uint32_t prefix = 0;
    uint32_t remaining = KK;
    #pragma unroll 1
    for (int pass = 0; pass < 4; ++pass) {
        const int      shift   = 24 - 8 * pass;
        const uint32_t hi_mask = (pass == 0) ? 0u : (0xFFFFFFFFu << (shift + 8));

        if (tid < 256) hist[tid] = 0u;
        __syncthreads();

        #pragma unroll 4
        for (int j = 0; j < EPT; ++j) {
            int idx = j * NTHREADS + tid;
            if (idx == target) continue;               // positive excluded
            uint32_t key = f2key(rowData[idx]);
            if ((key & hi_mask) == (prefix & hi_mask)) {
                atomicAdd(&hist[(key >> shift) & 0xFFu], 1u);   // ds_add_u32
            }
        }
        __syncthreads();

        if (tid == 0) {
            uint32_t cum = 0;
            uint32_t sel = 0;
            uint32_t rem = remaining;
            for (int i = 255; i >= 0; --i) {
                uint32_t c = hist[i];
                if (cum + c >= remaining) { sel = (uint32_t)i; rem = remaining - cum; break; }
                cum += c;
            }
            s_prefix    = prefix | (sel << shift);
            s_remaining = rem;
        }
        __syncthreads();
        prefix    = s_prefix;
        remaining = s_remaining;
        __syncthreads();
    }
    const uint32_t kth_key = prefix;     // exact threshold key
    const uint32_t ties    = remaining;  // # of threshold-valued entries kept

    // ---------------- row max (for stable logsumexp) -------------------------
    float lmax = -INFINITY;
    #pragma unroll 4
    for (int j = 0; j < EPT; ++j) {
        lmax = fmaxf(lmax, rowData[j * NTHREADS + tid]);   // includes positive (it's in the row)
    }
    red[tid] = lmax;
    __syncthreads();
    for (int s = NTHREADS / 2; s > 0; s >>= 1) {
        if (tid < s) red[tid] = fmaxf(red[tid], red[tid + s]);
        __syncthreads();
    }
    const float M = red[0];
    __syncthreads();

    // ---------------- sum of exp over strictly-above-threshold negatives -----
    float lsum = 0.0f;
    #pragma unroll 4
    for (int j = 0; j < EPT; ++j) {
        int idx = j * NTHREADS + tid;
        if (idx == target) continue;
        float v = rowData[idx];
        if (f2key(v) > kth_key) lsum += expf(SCALEF * (v - M));
    }
    red[tid] = lsum;
    __syncthreads();
    for (int s = NTHREADS / 2; s > 0; s >>= 1) {
        if (tid < s) red[tid] += red[tid + s];
        __syncthreads();
    }

    if (tid == 0) {
        float pos   = rowData[target];
        float vth   = key2f(kth_key);
        float total = red[0]
                    + (float)ties * expf(SCALEF * (vth - M))
                    + expf(SCALEF * (pos - M));
        row_loss[row] = logf(total) + SCALEF * M - SCALEF * pos;
    }
}

__global__ __launch_bounds__(NTHREADS, 1)
void reduce_mean_kernel(const float* __restrict__ row_loss, float* __restrict__ out) {
    __shared__ float red[NTHREADS];
    int tid = threadIdx.x;
    red[tid] = row_loss[tid] + row_loss[tid + NTHREADS];   // BB == 2*NTHREADS
    __syncthreads();
    for (int s = NTHREADS / 2; s > 0; s >>= 1) {
        if (tid < s) red[tid] += red[tid + s];
        __syncthreads();
    }
    if (tid == 0) out[0] = red[0] * (1.0f / (float)BB);
}

extern "C" void kernel_launch(void* const* d_in, const int* in_sizes, int n_in,
                              void* d_out, int out_size, void* d_ws, size_t ws_size,
                              hipStream_t stream) {
    (void)in_sizes; (void)n_in; (void)out_size; (void)ws_size;
    const float* logits  = (const float*)d_in[0];
    const int*   targets = (const int*)d_in[1];
    float*       out     = (float*)d_out;
    float*       rowloss = (float*)d_ws;   // BB floats of scratch

    hardneg_ce_rows<<<BB, NTHREADS, 0, stream>>>(logits, targets, rowloss);
    reduce_mean_kernel<<<1, NTHREADS, 0, stream>>>(rowloss, out);
}